// MHA_7121055777043
// MI455X (gfx1250) — compile-verified
//
#include <hip/hip_runtime.h>

// Linformer attention for MI455X (gfx1250), exact-fp32 path on
// V_WMMA_F32_16X16X4_F32 for every GEMM stage.
//
// Fragment layouts (CDNA5 ISA 7.12.2):
//   A (16x4 f32): lane L row m=L%16; vgpr j holds K = j + 2*(L/16)  -> b64 pair
//   B (4x16 f32): lane L col n=L%16; vgpr j holds K-row = j + 2*(L/16)
//     -> stored TRANSPOSED in LDS so the pair is a contiguous b64 load
//   C/D (16x16 f32): vgpr r: lanes 0-15 (M=r), lanes 16-31 (M=r+8), N=lane%16

typedef __attribute__((ext_vector_type(2))) float v2f;
typedef __attribute__((ext_vector_type(4))) float v4f;
typedef __attribute__((ext_vector_type(8))) float v8f;
typedef __attribute__((ext_vector_type(4))) int v4i;

__device__ __forceinline__ v8f wmma4(v2f a, v2f b, v8f c) {
  // 8 args: (neg_a, A, neg_b, B, c_mod, C, reuse_a, reuse_b)
  return __builtin_amdgcn_wmma_f32_16x16x4_f32(false, a, false, b, (short)0, c,
                                               false, false);
}

// ---- async global->LDS (CDNA5 GLOBAL_LOAD_ASYNC_TO_LDS_B128, ASYNCcnt) ----
#if __has_builtin(__builtin_amdgcn_global_load_async_to_lds_b128)
#define HAVE_ASYNC_LDS 1
typedef __attribute__((address_space(1))) v4i gv4i;  // global int4
typedef __attribute__((address_space(3))) v4i lv4i;  // LDS int4
#else
#define HAVE_ASYNC_LDS 0
#endif

__device__ __forceinline__ void copy16_g2lds(void* lds_dst, const void* gsrc) {
#if HAVE_ASYNC_LDS
  __builtin_amdgcn_global_load_async_to_lds_b128((gv4i*)gsrc, (lv4i*)lds_dst,
                                                 0, 0);
#else
  *(v4f*)lds_dst = *(const v4f*)gsrc;
#endif
}

__device__ __forceinline__ void async_join() {
#if HAVE_ASYNC_LDS
#if __has_builtin(__builtin_amdgcn_s_wait_asynccnt)
  __builtin_amdgcn_s_wait_asynccnt(0);
#else
  asm volatile("s_wait_asynccnt 0x0" ::: "memory");
#endif
#endif
}

// ---------------------------------------------------------------------------
// Batched GEMM: C[b] = A[b] (MxKc) * B[b] (KcxN), row-major, fp32.
// Grid: (N/BN, M/BM, batches). Block: 256 threads = 8 waves (4 M x 2 N).
// Per-wave micro-tile: (BM/4) x 64. BM=128 -> 8 accumulators, B-frag reuse x2.
// ---------------------------------------------------------------------------
constexpr int BN = 128, KC = 32;
constexpr int ASTR = KC + 4;  // 36: row stride mult-of-4 floats (b128 stores)
constexpr int BSTR = KC + 2;  // 34: even -> 8B-aligned b64 fragment pairs

template <int BM>
__global__ __launch_bounds__(256) void gemm_f32_wmma(
    const float* __restrict__ A, long long sA, int lda,
    const float* __restrict__ B, long long sB, int ldb,
    float* __restrict__ C, long long sC, int ldc, int Kc) {
  constexpr int WR = BM / 4;   // rows per wave: 16 or 32
  constexpr int NA = WR / 16;  // accumulator row-tiles per wave: 1 or 2
  __shared__ __align__(16) float As[BM][ASTR];
  __shared__ __align__(16) float Bst[BN][BSTR];  // transposed: [n][k]

  const long long bz = blockIdx.z;
  A += bz * sA;
  B += bz * sB;
  C += bz * sC;
  const int m0 = blockIdx.y * BM;
  const int n0 = blockIdx.x * BN;
  const int tid = threadIdx.x;
  const int lane = tid & 31, wave = tid >> 5;
  const int wm = wave & 3, wn = wave >> 2;
  const int half = lane >> 4, lm = lane & 15;

  v8f acc[NA][4] = {};

  for (int kc = 0; kc < Kc; kc += KC) {
    // A tile (BM x 32): straight copy -> async b128 path
    for (int i = tid; i < (BM * KC) / 4; i += 256) {
      const int r = i >> 3, c = (i & 7) << 2;
      copy16_g2lds(&As[r][c], &A[(long long)(m0 + r) * lda + kc + c]);
    }
    // B tile (32 x 128): transpose at store time so frags are b64-contiguous
    for (int i = tid; i < (KC * BN) / 4; i += 256) {
      const int r = i >> 5, c = (i & 31) << 2;
      const v4f bv = *(const v4f*)&B[(long long)(kc + r) * ldb + n0 + c];
      Bst[c + 0][r] = bv.x;
      Bst[c + 1][r] = bv.y;
      Bst[c + 2][r] = bv.z;
      Bst[c + 3][r] = bv.w;
    }
    async_join();
    __syncthreads();

#pragma unroll
    for (int kb = 0; kb < KC; kb += 4) {
      v2f a[NA];
#pragma unroll
      for (int i = 0; i < NA; ++i)
        a[i] = *(const v2f*)&As[wm * WR + i * 16 + lm][kb + 2 * half];
#pragma unroll
      for (int t = 0; t < 4; ++t) {
        const v2f bb = *(const v2f*)&Bst[wn * 64 + t * 16 + lm][kb + 2 * half];
#pragma unroll
        for (int i = 0; i < NA; ++i) acc[i][t] = wmma4(a[i], bb, acc[i][t]);
      }
    }
    __syncthreads();
  }

#pragma unroll
  for (int i = 0; i < NA; ++i)
#pragma unroll
    for (int t = 0; t < 4; ++t)
#pragma unroll
      for (int r = 0; r < 8; ++r) {
        const int row = m0 + wm * WR + i * 16 + r + 8 * half;
        const int col = n0 + wn * 64 + t * 16 + lm;
        C[(long long)row * ldc + col] = acc[i][t][r];
      }
}

// ---------------------------------------------------------------------------
// Fused attention per (b, h, 128-row strip): scores = Q*K^T, tril mask,
// softmax over K=64, out = attn*V.  8 waves x 16 query rows.
// qa[] holds the Q strip during the scores phase, then is recycled as the
// per-wave attention-weight tile (each wave touches only its own 16 rows).
// ---------------------------------------------------------------------------
constexpr float NEGV = -9e15f;
constexpr int QSTR = 68;  // mult-of-4 row stride (b128 stores, b64 frag reads)
constexpr int VSTR = 66;  // even row stride (b64 frag reads)

__global__ __launch_bounds__(256) void attn_f32_wmma(
    const float* __restrict__ q,  // [B*S, 1024] = [b,s,h,hd]
    const float* __restrict__ k,  // [B*K, 1024] = [b,kk,h,hd]
    const float* __restrict__ v,  // [B*K, 1024]
    float* __restrict__ out) {    // [B*S, 1024]
  __shared__ __align__(16) float qa[128][QSTR];  // Q strip, then attn weights
  __shared__ __align__(16) float ks[64][QSTR];   // K_h [kk][hd]
  __shared__ __align__(16) float vst[64][VSTR];  // V_h transposed [hd][kk]

  const int strip = blockIdx.x;  // 0..3
  const int h = blockIdx.y;      // 0..15
  const int b = blockIdx.z;      // 0..31
  const int tid = threadIdx.x;
  const int lane = tid & 31, wave = tid >> 5;
  const int half = lane >> 4, lm = lane & 15;

  // Stage K (async), V (transposed), Q strip (async).
  for (int i = tid; i < 64 * 16; i += 256) {  // 64 rows x 16 v4f
    const int r = i >> 4, c = (i & 15) << 2;
    const long long src = (long long)(b * 64 + r) * 1024 + h * 64 + c;
    copy16_g2lds(&ks[r][c], &k[src]);
    const v4f vv = *(const v4f*)&v[src];
    vst[c + 0][r] = vv.x;
    vst[c + 1][r] = vv.y;
    vst[c + 2][r] = vv.z;
    vst[c + 3][r] = vv.w;
  }
  for (int i = tid; i < 128 * 16; i += 256) {  // 128 rows x 16 v4f
    const int r = i >> 4, c = (i & 15) << 2;
    copy16_g2lds(&qa[r][c],
                 &q[(long long)(b * 512 + strip * 128 + r) * 1024 + h * 64 + c]);
  }
  async_join();
  __syncthreads();

  const int srow0 = strip * 128 + wave * 16;

  // scores = Q * K^T (contraction over HD=64); both frags are b64 LDS loads
  v8f sc[4] = {};
#pragma unroll
  for (int kb = 0; kb < 64; kb += 4) {
    const v2f a = *(const v2f*)&qa[wave * 16 + lm][kb + 2 * half];
#pragma unroll
    for (int t = 0; t < 4; ++t) {
      const v2f bb = *(const v2f*)&ks[t * 16 + lm][kb + 2 * half];
      sc[t] = wmma4(a, bb, sc[t]);
    }
  }

  // tril mask + row softmax. Row m = r + 8*half lives in one 16-lane group;
  // each lane holds 4 columns (16t+lm); xor masks 1/2/4/8 stay in-group.
#pragma unroll
  for (int r = 0; r < 8; ++r) {
    const int srow = srow0 + r + 8 * half;
    float vals[4];
    float mx = NEGV;
#pragma unroll
    for (int t = 0; t < 4; ++t) {
      const int col = t * 16 + lm;
      float x = sc[t][r];
      x = (col <= srow) ? x : NEGV;
      vals[t] = x;
      mx = fmaxf(mx, x);
    }
    mx = fmaxf(mx, __shfl_xor(mx, 1));
    mx = fmaxf(mx, __shfl_xor(mx, 2));
    mx = fmaxf(mx, __shfl_xor(mx, 4));
    mx = fmaxf(mx, __shfl_xor(mx, 8));
    float s = 0.f;
#pragma unroll
    for (int t = 0; t < 4; ++t) {
      vals[t] = __expf(vals[t] - mx);  // masked entries underflow to 0
      s += vals[t];
    }
    s += __shfl_xor(s, 1);
    s += __shfl_xor(s, 2);
    s += __shfl_xor(s, 4);
    s += __shfl_xor(s, 8);
    const float inv = 1.0f / s;
    // Recycle qa: this wave's Q rows are fully consumed (same-wave ds order).
#pragma unroll
    for (int t = 0; t < 4; ++t)
      qa[wave * 16 + r + 8 * half][t * 16 + lm] = vals[t] * inv;
  }

  // out = attn * V (contraction over K=64); V read via transposed tile
  v8f ov[4] = {};
#pragma unroll
  for (int kb = 0; kb < 64; kb += 4) {
    const v2f a = *(const v2f*)&qa[wave * 16 + lm][kb + 2 * half];
#pragma unroll
    for (int t = 0; t < 4; ++t) {
      const v2f bb = *(const v2f*)&vst[t * 16 + lm][kb + 2 * half];
      ov[t] = wmma4(a, bb, ov[t]);
    }
  }

  float* orow = &out[(long long)(b * 512 + srow0) * 1024 + h * 64];
#pragma unroll
  for (int t = 0; t < 4; ++t)
#pragma unroll
    for (int r = 0; r < 8; ++r)
      orow[(long long)(r + 8 * half) * 1024 + t * 16 + lm] = ov[t][r];
}

// ---------------------------------------------------------------------------
extern "C" void kernel_launch(void* const* d_in, const int* in_sizes, int n_in,
                              void* d_out, int out_size, void* d_ws,
                              size_t ws_size, hipStream_t stream) {
  const float* query = (const float*)d_in[0];  // [32,512,1024]
  const float* key   = (const float*)d_in[1];  // [32,512,1024]
  const float* value = (const float*)d_in[2];  // [32,512,1024]
  const float* Wq    = (const float*)d_in[3];  // [1024,16,64]
  const float* Wk    = (const float*)d_in[4];
  const float* Wv    = (const float*)d_in[5];
  const float* Kd    = (const float*)d_in[6];  // [64,512]
  const float* Vd    = (const float*)d_in[7];  // [64,512]
  float* out = (float*)d_out;                  // [32,512,1024]

  // Workspace (fp32 elements), 96 MB total.
  float* ws = (float*)d_ws;
  float* key_ds = ws;                               // 32*64*1024
  float* val_ds = key_ds + (size_t)32 * 64 * 1024;  // 32*64*1024
  float* qb     = val_ds + (size_t)32 * 64 * 1024;  // 32*512*1024
  float* kb     = qb + (size_t)32 * 512 * 1024;     // 32*64*1024
  float* vb     = kb + (size_t)32 * 64 * 1024;      // 32*64*1024

  const long long sKV = 512LL * 1024, sDS = 64LL * 1024;

  // 1) Linformer downsampling: [64,512] x [512,1024] per batch (M=64 -> BM=64)
  gemm_f32_wmma<64><<<dim3(1024 / BN, 1, 32), 256, 0, stream>>>(
      Kd, 0, 512, key, sKV, 1024, key_ds, sDS, 1024, 512);
  gemm_f32_wmma<64><<<dim3(1024 / BN, 1, 32), 256, 0, stream>>>(
      Vd, 0, 512, value, sKV, 1024, val_ds, sDS, 1024, 512);

  // 2) Head projections (flattened single GEMMs, BM=128)
  gemm_f32_wmma<128><<<dim3(1024 / BN, 16384 / 128, 1), 256, 0, stream>>>(
      query, 0, 1024, Wq, 0, 1024, qb, 0, 1024, 1024);
  gemm_f32_wmma<128><<<dim3(1024 / BN, 2048 / 128, 1), 256, 0, stream>>>(
      key_ds, 0, 1024, Wk, 0, 1024, kb, 0, 1024, 1024);
  gemm_f32_wmma<128><<<dim3(1024 / BN, 2048 / 128, 1), 256, 0, stream>>>(
      val_ds, 0, 1024, Wv, 0, 1024, vb, 0, 1024, 1024);

  // 3) Fused masked-softmax attention
  attn_f32_wmma<<<dim3(512 / 128, 16, 32), 256, 0, stream>>>(qb, kb, vb, out);
}